// FcosDetector_46127948759499
// MI455X (gfx1250) — compile-verified
//
#include <hip/hip_runtime.h>
#include <hip/hip_bf16.h>
#include <math.h>

typedef __attribute__((ext_vector_type(2))) float v2f;
typedef __attribute__((ext_vector_type(8))) float v8f;

#define NPOS 27280
#define KPAD 1024
#define NKEEP 1000
#define NBATCH 16
#define NCLS 20

// ---------------------------------------------------------------------------
// Kernel 1: per-level decode. score = sigmoid(max_c logit), kind = argmax_c,
// boxes from exp(reg) around stride-scaled centers.
// ---------------------------------------------------------------------------
__global__ void decode_kernel(const float* __restrict__ cls,
                              const float* __restrict__ reg,
                              float* __restrict__ score, int* __restrict__ kind,
                              float* __restrict__ boxes,
                              int H, int W, float stride, int base) {
  int idx = blockIdx.x * blockDim.x + threadIdx.x;
  int HW = H * W;
  int total = NBATCH * HW;
  if (idx >= total) return;
  int b = idx / HW;
  int pos = idx - b * HW;
  int h = pos / W;
  int w = pos - h * W;

  const float* cb = cls + (size_t)b * NCLS * HW + pos;
  const float* rb = reg + (size_t)b * 4 * HW + pos;
  __builtin_prefetch(rb, 0, 1);  // gfx1250: global_prefetch_b8

  float maxv = cb[0];
  int argc = 0;
#pragma unroll
  for (int c = 1; c < NCLS; ++c) {
    float v = cb[(size_t)c * HW];
    if (v > maxv) { maxv = v; argc = c; }
  }
  float sc = 1.0f / (1.0f + expf(-maxv));

  float dl = expf(rb[0]);
  float dt = expf(rb[(size_t)1 * HW]);
  float dr = expf(rb[(size_t)2 * HW]);
  float db = expf(rb[(size_t)3 * HW]);
  float cx = ((float)w + 0.5f) * stride;
  float cy = ((float)h + 0.5f) * stride;

  size_t o = (size_t)b * NPOS + base + pos;
  score[o] = sc;
  kind[o] = argc;
  float* bo = boxes + o * 4;
  bo[0] = cx - dl; bo[1] = cy - dt; bo[2] = cx + dr; bo[3] = cy + db;
}

// ---------------------------------------------------------------------------
// Kernel 2: per-batch exact top-1000 via 46-bit radix select + bitonic sort.
// key = (float_bits(score) << 15) | (32767 - idx)  -> unique, top_k-compatible.
// ---------------------------------------------------------------------------
__device__ __forceinline__ unsigned long long mk_key(float s, int idx) {
  unsigned int b = __float_as_uint(s);  // scores > 0 => uint order == value order
  return ((unsigned long long)b << 15) | (unsigned long long)(32767 - idx);
}

__global__ void __launch_bounds__(1024) topk_kernel(
    const float* __restrict__ score, const int* __restrict__ kind,
    const float* __restrict__ boxes, float* __restrict__ topS,
    float* __restrict__ topBox, int* __restrict__ topKind) {
  int b = blockIdx.x;
  int tid = threadIdx.x;
  const float* sc = score + (size_t)b * NPOS;

  __shared__ unsigned int hist[4096];
  __shared__ unsigned long long P;      // accumulated threshold prefix
  __shared__ int needSh;
  __shared__ unsigned int cnt;
  __shared__ float sS[KPAD];
  __shared__ int sI[KPAD];

  if (tid == 0) { P = 0ULL; needSh = NKEEP; cnt = 0u; }
  __syncthreads();

  const int shifts[4] = {34, 22, 10, 0};
  const int widths[4] = {12, 12, 12, 10};
#pragma unroll 1
  for (int pass = 0; pass < 4; ++pass) {
    int shift = shifts[pass];
    int bins = 1 << widths[pass];
    unsigned long long hiMask = ~((1ULL << (shift + widths[pass])) - 1ULL);
    for (int i = tid; i < bins; i += 1024) hist[i] = 0u;
    __syncthreads();
    unsigned long long Ploc = P;
    for (int i = tid; i < NPOS; i += 1024) {
      unsigned long long k = mk_key(sc[i], i);
      if ((k & hiMask) == Ploc)
        atomicAdd(&hist[(unsigned)((k >> shift) & (unsigned long long)(bins - 1))], 1u);
    }
    __syncthreads();
    if (tid == 0) {
      int need = needSh;
      unsigned int cum = 0;
      int d = bins - 1;
      for (; d > 0; --d) {
        if (cum + hist[d] >= (unsigned)need) break;
        cum += hist[d];
      }
      P |= ((unsigned long long)d) << shift;
      needSh = need - (int)cum;
    }
    __syncthreads();
  }
  unsigned long long T = P;  // exact key of the 1000th-largest element

  // deterministic-set compaction (order fixed later by sort; keys unique)
  sS[tid] = -INFINITY;
  sI[tid] = 0x7FFFFFFF;
  __syncthreads();
  for (int i = tid; i < NPOS; i += 1024) {
    float s = sc[i];
    if (mk_key(s, i) >= T) {
      unsigned int p = atomicAdd(&cnt, 1u);
      if (p < KPAD) { sS[p] = s; sI[p] = i; }
    }
  }
  __syncthreads();

  // bitonic sort 1024 elems, descending by (score, then lowest index)
  for (int k = 2; k <= KPAD; k <<= 1) {
    for (int j = k >> 1; j > 0; j >>= 1) {
      int ixj = tid ^ j;
      if (ixj > tid) {
        float s1 = sS[tid], s2 = sS[ixj];
        int i1 = sI[tid], i2 = sI[ixj];
        bool firstGreater = (s1 > s2) || (s1 == s2 && i1 < i2);
        bool descRegion = (tid & k) == 0;
        if (descRegion ? !firstGreater : firstGreater) {
          sS[tid] = s2; sS[ixj] = s1;
          sI[tid] = i2; sI[ixj] = i1;
        }
      }
      __syncthreads();
    }
  }

  // gather
  int kk = tid;
  float s = sS[kk];
  int i = sI[kk];
  size_t ob = (size_t)b * KPAD + kk;
  topS[ob] = s;
  bool valid = (kk < NKEEP) && (i < NPOS);
  float bx0 = 0.f, bx1 = 0.f, bx2 = 0.f, bx3 = 0.f;
  int kd = 0;
  if (valid) {
    const float* bp = boxes + ((size_t)b * NPOS + i) * 4;
    bx0 = bp[0]; bx1 = bp[1]; bx2 = bp[2]; bx3 = bp[3];
    kd = kind[(size_t)b * NPOS + i];
  }
  float* tb = topBox + ob * 4;
  tb[0] = bx0; tb[1] = bx1; tb[2] = bx2; tb[3] = bx3;
  topKind[ob] = kd;
}

// ---------------------------------------------------------------------------
// Kernel 3: NMS map. One wave per 16x16 IoU tile. The pairwise denominator
// area_i + area_j is a rank-2 matmul computed with V_WMMA_F32_16X16X4_F32:
//   A(16x4) = [area_i, 1, 0, 0],  B(4x16) = [ones; area_j; 0; 0]
// A layout (ISA 7.12.2): lanes0-15 V0=K0,V1=K1; lanes16-31 V0=K2,V1=K3.
// B follows the same v / v+half interleave as C/D.
// Suppression bits (iou>0.5 && j>i) packed per row via wave32 ballot.
// ---------------------------------------------------------------------------
__global__ void __launch_bounds__(256) nms_map_kernel(
    const float* __restrict__ topBox, unsigned short* __restrict__ supMat) {
  int wave = threadIdx.x >> 5;
  int lane = threadIdx.x & 31;
  int tileG = blockIdx.x * 8 + wave;         // total tiles = NBATCH*64*64
  int b = tileG >> 12;
  int t = tileG & 4095;
  int ti = t >> 6, tj = t & 63;
  int i0 = ti << 4, j0 = tj << 4;
  const float* Bb = topBox + (size_t)b * KPAD * 4;

  int nlo = lane & 15;
  bool lo = lane < 16;

  // j-side box for this lane's column (both halves read the same 16 rows)
  const float* bj = Bb + (size_t)(j0 + nlo) * 4;
  float jx1 = bj[0], jy1 = bj[1], jx2 = bj[2], jy2 = bj[3];
  float jArea = fmaxf(jx2 - jx1, 0.f) * fmaxf(jy2 - jy1, 0.f);

  // i-side area for A operand
  const float* bia = Bb + (size_t)(i0 + nlo) * 4;
  float aArea = fmaxf(bia[2] - bia[0], 0.f) * fmaxf(bia[3] - bia[1], 0.f);

  v2f A, Bm;
  A[0] = lo ? aArea : 0.f;   // K=0 row contribution: area_i   (K=2 is zero)
  A[1] = lo ? 1.0f : 0.f;    // K=1 row contribution: 1        (K=3 is zero)
  Bm[0] = lo ? 1.0f : 0.f;   // B row0 = ones                  (row2 zero)
  Bm[1] = lo ? jArea : 0.f;  // B row1 = area_j                (row3 zero)
  v8f Cz = {};
  v8f D = __builtin_amdgcn_wmma_f32_16x16x4_f32(
      false, A, false, Bm, (short)0, Cz, false, false);  // D[m][n] = area_i+area_j

  int half = lo ? 0 : 8;
  int gj = j0 + nlo;
#pragma unroll
  for (int v = 0; v < 8; ++v) {
    int gi = i0 + v + half;
    const float* bi = Bb + (size_t)gi * 4;
    float xx1 = fmaxf(bi[0], jx1);
    float yy1 = fmaxf(bi[1], jy1);
    float xx2 = fminf(bi[2], jx2);
    float yy2 = fminf(bi[3], jy2);
    float inter = fmaxf(xx2 - xx1, 0.f) * fmaxf(yy2 - yy1, 0.f);
    float iou = inter / (D[v] - inter + 1e-9f);
    bool pred = (iou > 0.5f) && (gj > gi);
    unsigned int m = __builtin_amdgcn_ballot_w32(pred);
    if (lane == 0)
      supMat[((size_t)b * KPAD + i0 + v) * 64 + tj] = (unsigned short)(m & 0xFFFFu);
    if (lane == 16)
      supMat[((size_t)b * KPAD + i0 + v + 8) * 64 + tj] = (unsigned short)(m >> 16);
  }
}

// ---------------------------------------------------------------------------
// Kernel 4: sequential NMS reduce (exact reference recursion) + output pack.
// 64 threads hold the 1024-bit keep mask in LDS (16 bits per thread-word).
// ---------------------------------------------------------------------------
__global__ void nms_reduce_kernel(const float* __restrict__ topS,
                                  const float* __restrict__ topBox,
                                  const int* __restrict__ topKind,
                                  const unsigned short* __restrict__ supMat,
                                  float* __restrict__ out) {
  int b = blockIdx.x;
  int tid = threadIdx.x;  // 64 threads
  __shared__ unsigned int keepW[64];

  unsigned int bits = 0;
  for (int t = 0; t < 16; ++t) {
    int k = tid * 16 + t;
    if (k < NKEEP && topS[(size_t)b * KPAD + k] > 0.05f) bits |= (1u << t);
  }
  keepW[tid] = bits;
  __syncthreads();

  for (int i = 0; i < NKEEP; ++i) {
    bool ki = (keepW[i >> 4] >> (i & 15)) & 1u;
    __syncthreads();
    if (ki) {
      unsigned int sup = supMat[((size_t)b * KPAD + i) * 64 + tid];
      keepW[tid] &= ~sup;
    }
    __syncthreads();
  }

  for (int k = tid; k < NKEEP; k += blockDim.x) {
    size_t ib = (size_t)b * KPAD + k;
    size_t ob = ((size_t)b * NKEEP + k) * 7;
    const float* tb = topBox + ib * 4;
    out[ob + 0] = tb[0];
    out[ob + 1] = tb[1];
    out[ob + 2] = tb[2];
    out[ob + 3] = tb[3];
    out[ob + 4] = (float)topKind[ib];
    out[ob + 5] = topS[ib];
    out[ob + 6] = ((keepW[k >> 4] >> (k & 15)) & 1u) ? 1.0f : 0.0f;
  }
}

// ---------------------------------------------------------------------------
extern "C" void kernel_launch(void* const* d_in, const int* in_sizes, int n_in,
                              void* d_out, int out_size, void* d_ws, size_t ws_size,
                              hipStream_t stream) {
  (void)in_sizes; (void)n_in; (void)out_size; (void)ws_size;
  static const int Hs[5] = {128, 64, 32, 16, 8};
  static const int Ws[5] = {160, 80, 40, 20, 10};
  static const float Ss[5] = {8.f, 16.f, 32.f, 64.f, 128.f};

  char* p = (char*)d_ws;
  auto alloc = [&](size_t bytes) {
    char* r = p;
    p += (bytes + 255) & ~(size_t)255;
    return r;
  };
  float* score  = (float*)alloc((size_t)NBATCH * NPOS * sizeof(float));
  int*   kindA  = (int*)  alloc((size_t)NBATCH * NPOS * sizeof(int));
  float* boxes  = (float*)alloc((size_t)NBATCH * NPOS * 4 * sizeof(float));
  float* topS   = (float*)alloc((size_t)NBATCH * KPAD * sizeof(float));
  float* topBox = (float*)alloc((size_t)NBATCH * KPAD * 4 * sizeof(float));
  int*   topKind= (int*)  alloc((size_t)NBATCH * KPAD * sizeof(int));
  unsigned short* supMat =
      (unsigned short*)alloc((size_t)NBATCH * KPAD * 64 * sizeof(unsigned short));

  int base = 0;
  for (int L = 0; L < 5; ++L) {
    int H = Hs[L], W = Ws[L];
    int tot = NBATCH * H * W;
    decode_kernel<<<(tot + 255) / 256, 256, 0, stream>>>(
        (const float*)d_in[3 * L], (const float*)d_in[3 * L + 2],
        score, kindA, boxes, H, W, Ss[L], base);
    base += H * W;
  }

  topk_kernel<<<NBATCH, 1024, 0, stream>>>(score, kindA, boxes,
                                           topS, topBox, topKind);

  // NBATCH * 64 * 64 tiles, 8 waves (tiles) per 256-thread block
  nms_map_kernel<<<(NBATCH * 4096) / 8, 256, 0, stream>>>(topBox, supMat);

  nms_reduce_kernel<<<NBATCH, 64, 0, stream>>>(topS, topBox, topKind, supMat,
                                               (float*)d_out);
}